// EntmaxAlphaActivation_30477087933159
// MI455X (gfx1250) — compile-verified
//
#include <hip/hip_runtime.h>

// entmax-bisect (alpha-entmax) over last dim, S = 4096 fixed (matches reference).
// One 256-thread block per row; row cached in VGPRs; CDNA5 async global->LDS
// staging (global_load_async_to_lds_b128 th:TH_LOAD_NT / s_wait_asynccnt);
// single-barrier double-buffered block reductions; NT output stores.

#define S_DIM   4096
#define BLK     256
#define EPT     (S_DIM / BLK)   // 16 elements per thread
#define NWAVES  (BLK / 32)      // 8 wave32 per block
#define N_ITER  50
#define NEG_VAL (-1e9f)

__device__ __forceinline__ float wave_reduce_sum(float v) {
#pragma unroll
  for (int off = 16; off > 0; off >>= 1)
    v += __shfl_xor(v, off, 32);
  return v;   // butterfly: identical value in all 32 lanes
}

__device__ __forceinline__ float wave_reduce_max(float v) {
#pragma unroll
  for (int off = 16; off > 0; off >>= 1)
    v = fmaxf(v, __shfl_xor(v, off, 32));
  return v;
}

// Double-buffered, ONE barrier per call. Safe because a write to buffer P at
// call N+2 occurs after call N+1's barrier, which each wave signals only after
// finishing its call-N reads of buffer P. Callers must alternate parity.
__device__ __forceinline__ float block_reduce_sum_db(float v,
                                                     float (*s_red)[NWAVES],
                                                     int parity) {
  v = wave_reduce_sum(v);
  if ((threadIdx.x & 31) == 0) s_red[parity][threadIdx.x >> 5] = v;
  __syncthreads();
  float tot = 0.0f;
#pragma unroll
  for (int i = 0; i < NWAVES; ++i) tot += s_red[parity][i];  // same order everywhere
  return tot;
}

// Fully-barriered max reduce (called once; trailing barrier re-frees buffer 0).
__device__ __forceinline__ float block_reduce_max(float v,
                                                  float (*s_red)[NWAVES]) {
  v = wave_reduce_max(v);
  if ((threadIdx.x & 31) == 0) s_red[0][threadIdx.x >> 5] = v;
  __syncthreads();
  float m = -__builtin_inff();
#pragma unroll
  for (int i = 0; i < NWAVES; ++i) m = fmaxf(m, s_red[0][i]);
  __syncthreads();
  return m;
}

__global__ void __launch_bounds__(BLK)
entmax_bisect_kernel(const float* __restrict__ scores,
                     const unsigned char* __restrict__ mask,
                     const float* __restrict__ alpha_p,
                     float* __restrict__ out) {
  __shared__ float         s_x[S_DIM];          // 16 KB staged scores
  __shared__ unsigned char s_m[S_DIM];          //  4 KB staged mask (bool = u8)
  __shared__ float         s_red[2][NWAVES];    // double-buffered reduce scratch

  const int    t      = threadIdx.x;
  const size_t rowoff = (size_t)blockIdx.x * S_DIM;

  // ---- Stage row (scores + mask) into LDS with CDNA5 async loads, NT ------
  {
    unsigned long long gx = (unsigned long long)(const void*)(scores + rowoff);
    unsigned           lx = (unsigned)(size_t)(&s_x[0]);
#pragma unroll
    for (int p = 0; p < 4; ++p) {
      unsigned           off = (unsigned)(p * 4096 + t * 16);   // bytes
      unsigned           la  = lx + off;
      unsigned long long ga  = gx + off;
      asm volatile("global_load_async_to_lds_b128 %0, %1, off th:TH_LOAD_NT"
                   :: "v"(la), "v"(ga) : "memory");
    }
    unsigned           lm = (unsigned)(size_t)(&s_m[0]) + (unsigned)(t * 16);
    unsigned long long gm = (unsigned long long)(const void*)(mask + rowoff)
                            + (unsigned long long)(t * 16);
    asm volatile("global_load_async_to_lds_b128 %0, %1, off th:TH_LOAD_NT"
                 :: "v"(lm), "v"(gm) : "memory");
    asm volatile("s_wait_asynccnt 0" ::: "memory");
  }
  __syncthreads();

  // ---- Scalar parameters (uniform) ----------------------------------------
  float alpha = fmaxf(alpha_p[0], 1.0f);
  const float am1 = alpha - 1.0f;
  const float inv = 1.0f / am1;
  const bool  sq  = (inv == 2.0f);   // alpha == 1.5 fast path: p = t*t

  // ---- Pull this thread's 16 elements into VGPRs (conflict-free stride) ---
  float xs[EPT];
  float lmax = -__builtin_inff();
#pragma unroll
  for (int k = 0; k < EPT; ++k) {
    const int idx = k * BLK + t;
    float v = s_m[idx] ? s_x[idx] : NEG_VAL;
    v *= am1;
    xs[k] = v;
    lmax = fmaxf(lmax, v);
  }
  const float max_val = block_reduce_max(lmax, s_red);

  // ---- Bisection init ------------------------------------------------------
  float tau_lo = max_val - 1.0f;
  // (1/d)^am1 = exp2(am1 * log2(1/d)); d power of two -> exact
  const float pw = __builtin_amdgcn_exp2f(
      am1 * __builtin_amdgcn_logf(1.0f / (float)S_DIM));
  float dm    = (max_val - pw) - tau_lo;
  float tau_m = tau_lo;

  // ---- 50 bisection iterations (exactly as reference) ----------------------
  for (int it = 0; it < N_ITER; ++it) {
    dm *= 0.5f;
    tau_m = tau_lo + dm;
    float ls = 0.0f;
    if (sq) {
#pragma unroll
      for (int k = 0; k < EPT; ++k) {
        const float tv = fmaxf(xs[k] - tau_m, 0.0f);
        ls = __builtin_fmaf(tv, tv, ls);              // v_fmac_f32
      }
    } else {
#pragma unroll
      for (int k = 0; k < EPT; ++k) {
        const float tv = fmaxf(xs[k] - tau_m, 0.0f);  // log2(0) = -inf -> p = 0
        ls += __builtin_amdgcn_exp2f(inv * __builtin_amdgcn_logf(tv));
      }
    }
    const float fsum = block_reduce_sum_db(ls, s_red, it & 1);
    if (fsum - 1.0f >= 0.0f) tau_lo = tau_m;          // uniform branch
  }

  // ---- p from last tau_m, then normalize (ensure_sum_one) ------------------
  float p[EPT];
  float ls = 0.0f;
  if (sq) {
#pragma unroll
    for (int k = 0; k < EPT; ++k) {
      const float tv = fmaxf(xs[k] - tau_m, 0.0f);
      const float pv = tv * tv;
      p[k] = pv;
      ls += pv;
    }
  } else {
#pragma unroll
    for (int k = 0; k < EPT; ++k) {
      const float tv = fmaxf(xs[k] - tau_m, 0.0f);
      const float pv = __builtin_amdgcn_exp2f(inv * __builtin_amdgcn_logf(tv));
      p[k] = pv;
      ls += pv;
    }
  }
  // last loop call used parity 1 (it = 49), so parity 0 is free here
  const float tot  = block_reduce_sum_db(ls, s_red, N_ITER & 1 ? 1 : 0);
  const float rnrm = 1.0f / tot;
#pragma unroll
  for (int k = 0; k < EPT; ++k)
    __builtin_nontemporal_store(p[k] * rnrm, &out[rowoff + (size_t)(k * BLK + t)]);
}

extern "C" void kernel_launch(void* const* d_in, const int* in_sizes, int n_in,
                              void* d_out, int out_size, void* d_ws, size_t ws_size,
                              hipStream_t stream) {
  const float*         scores = (const float*)d_in[0];
  const unsigned char* mask   = (const unsigned char*)d_in[1]; // jnp bool = u8
  const float*         alpha  = (const float*)d_in[2];
  float*               out    = (float*)d_out;

  const int B = in_sizes[0] / S_DIM;   // rows
  entmax_bisect_kernel<<<dim3(B), dim3(BLK), 0, stream>>>(scores, mask, alpha, out);
}